// GATLinkPred_618475291071
// MI455X (gfx1250) — compile-verified
//
#include <hip/hip_runtime.h>
#include <hip/hip_bf16.h>

typedef __attribute__((ext_vector_type(2))) float v2f;
typedef __attribute__((ext_vector_type(8))) float v8f;

#define NEG_SLOPE 0.2f
#define EPSV 1e-16f

// ---------- helpers ----------

// order-preserving encode of float into uint so atomicMax works for all signs
__device__ __forceinline__ unsigned f2ord(float f) {
    unsigned u = __float_as_uint(f);
    return (u & 0x80000000u) ? ~u : (u | 0x80000000u);
}
__device__ __forceinline__ float ord2f(unsigned e) {
    return (e & 0x80000000u) ? __uint_as_float(e & 0x7FFFFFFFu)
                             : __uint_as_float(~e);
}

// edge list with appended self loops: i < E -> (src,dst) from edge_index, else node i-E
__device__ __forceinline__ void edge_nodes(const long long* __restrict__ ei,
                                           int E, int i, int& s, int& d) {
    if (i < E) { s = (int)ei[i]; d = (int)ei[E + i]; }
    else       { s = d = i - E; }
}

// ---------- fill ----------

__global__ void fill_u32_kernel(unsigned* __restrict__ p, unsigned v, long long n) {
    long long i = (long long)blockIdx.x * blockDim.x + threadIdx.x;
    if (i < n) p[i] = v;
}

// ---------- fp32 WMMA GEMM: C[M,Nc] = A[M,K] * B[K,Nc], all row-major ----------
// One wave32 computes one 16x16 output tile with V_WMMA_F32_16X16X4_F32 over K/4 steps.
// ISA VGPR layouts (CDNA5 ISA):
//   A 16x4 f32 : lane<16 -> M=lane,   VGPR0=K0,VGPR1=K1 ; lane>=16 -> M=lane-16, K2/K3
//   B 4x16 f32 : lane<16 -> N=lane,   VGPR0=K0,VGPR1=K1 ; lane>=16 -> N=lane-16, K2/K3
//   C 16x16 f32: VGPR r, lane l -> row = r + 8*(l>>4), col = l&15
__global__ __launch_bounds__(256) void wmma_gemm_f32_kernel(
    const float* __restrict__ A, const float* __restrict__ B, float* __restrict__ C,
    int M, int K, int Nc) {
    const int lane  = threadIdx.x & 31;
    const int wave  = threadIdx.x >> 5;
    const int tilesN = Nc >> 4;
    const int t = blockIdx.x * 8 + wave;             // wave-uniform tile id
    const int totalTiles = (M >> 4) * tilesN;
    if (t >= totalTiles) return;                     // wave-uniform exit (EXEC stays all-1 for WMMA)
    const int tm = t / tilesN;
    const int tn = t - tm * tilesN;
    const int half = lane >> 4;                      // 0: K pair {0,1}; 1: K pair {2,3}
    const int r    = lane & 15;

    const float* Arow = A + (size_t)(tm * 16 + r) * K;
    const float* Bcol = B + (size_t)(tn * 16 + r);

    v8f acc = {0.f, 0.f, 0.f, 0.f, 0.f, 0.f, 0.f, 0.f};
    for (int kb = 0; kb < K; kb += 4) {
        const int k0 = kb + half * 2;
        v2f av, bv;
        av.x = Arow[k0];
        av.y = Arow[k0 + 1];                         // contiguous pair -> global_load_b64
        bv.x = Bcol[(size_t)k0 * Nc];
        bv.y = Bcol[(size_t)(k0 + 1) * Nc];
        acc = __builtin_amdgcn_wmma_f32_16x16x4_f32(
            /*neg_a=*/false, av, /*neg_b=*/false, bv,
            /*c_mod=*/(short)0, acc, /*reuse_a=*/false, /*reuse_b=*/false);
    }
#pragma unroll
    for (int i = 0; i < 8; ++i) {
        const int row = tm * 16 + i + half * 8;
        C[(size_t)row * Nc + tn * 16 + r] = acc[i];
    }
}

// ---------- per-(node,head) attention logits: one wave32 each ----------
// asrc[n,h] = sum_c h[n,h,c]*a_src[h,c] ; adst likewise
__global__ void alpha_kernel(const float* __restrict__ h,
                             const float* __restrict__ a_src,
                             const float* __restrict__ a_dst,
                             float* __restrict__ asrc, float* __restrict__ adst,
                             int N, int H, int C) {
    const int w    = (blockIdx.x * blockDim.x + threadIdx.x) >> 5;
    const int lane = threadIdx.x & 31;
    if (w >= N * H) return;
    const int node = w / H;
    const int head = w - node * H;
    const float* hv = h + ((size_t)node * H + head) * C;
    const float* as = a_src + (size_t)head * C;
    const float* ad = a_dst + (size_t)head * C;
    float s0 = 0.f, s1 = 0.f;
    for (int c = lane * 4; c < C; c += 128) {        // 32 lanes * float4
        const float4 x4 = *(const float4*)(hv + c);
        const float4 a4 = *(const float4*)(as + c);
        const float4 d4 = *(const float4*)(ad + c);
        s0 += x4.x * a4.x + x4.y * a4.y + x4.z * a4.z + x4.w * a4.w;
        s1 += x4.x * d4.x + x4.y * d4.y + x4.z * d4.z + x4.w * d4.w;
    }
#pragma unroll
    for (int off = 16; off > 0; off >>= 1) {
        s0 += __shfl_down(s0, off, 32);
        s1 += __shfl_down(s1, off, 32);
    }
    if (lane == 0) { asrc[w] = s0; adst[w] = s1; }
}

// ---------- edge passes ----------

__global__ void edge_max_kernel(const float* __restrict__ asrc, const float* __restrict__ adst,
                                const long long* __restrict__ ei, unsigned* __restrict__ m,
                                int E, int Etot, int H) {
    const int idx = blockIdx.x * blockDim.x + threadIdx.x;
    if (idx >= Etot * H) return;
    const int edge = idx / H, head = idx - edge * H;
    int s, d; edge_nodes(ei, E, edge, s, d);
    float e = asrc[(size_t)s * H + head] + adst[(size_t)d * H + head];
    e = (e < 0.f) ? NEG_SLOPE * e : e;
    atomicMax(&m[(size_t)d * H + head], f2ord(e));
}

__global__ void decode_max_kernel(unsigned* __restrict__ m, int n) {
    const int i = blockIdx.x * blockDim.x + threadIdx.x;
    if (i < n) {
        const float f = ord2f(m[i]);
        ((float*)m)[i] = f;
    }
}

__global__ void edge_exp_kernel(const float* __restrict__ asrc, const float* __restrict__ adst,
                                const long long* __restrict__ ei,
                                const float* __restrict__ mflt, float* __restrict__ ssum,
                                float* __restrict__ exb, int E, int Etot, int H) {
    const int idx = blockIdx.x * blockDim.x + threadIdx.x;
    if (idx >= Etot * H) return;
    const int edge = idx / H, head = idx - edge * H;
    int s, d; edge_nodes(ei, E, edge, s, d);
    float e = asrc[(size_t)s * H + head] + adst[(size_t)d * H + head];
    e = (e < 0.f) ? NEG_SLOPE * e : e;
    const float ex = __expf(e - mflt[(size_t)d * H + head]);
    exb[idx] = ex;
    atomicAdd(&ssum[(size_t)d * H + head], ex);
}

__global__ void edge_norm_kernel(const long long* __restrict__ ei,
                                 const float* __restrict__ ssum, float* __restrict__ exb,
                                 int E, int Etot, int H) {
    const int idx = blockIdx.x * blockDim.x + threadIdx.x;
    if (idx >= Etot * H) return;
    const int edge = idx / H, head = idx - edge * H;
    int s, d; edge_nodes(ei, E, edge, s, d);
    exb[idx] = exb[idx] / (ssum[(size_t)d * H + head] + EPSV);
}

// scatter-aggregate: out[dst] += h[src] * alpha ; HC/4 threads per edge, float4 each
__global__ void agg_kernel(const float* __restrict__ h, const float* __restrict__ alpha,
                           const long long* __restrict__ ei, float* __restrict__ out,
                           int E, int Etot, int H, int C, int HC) {
    const int tpe = HC >> 2;
    const long long idx = (long long)blockIdx.x * blockDim.x + threadIdx.x;
    const long long edge = idx / tpe;
    const int t = (int)(idx - edge * (long long)tpe);
    if (edge >= Etot) return;
    int s, d; edge_nodes(ei, E, (int)edge, s, d);
    const int c0 = t * 4;
    const int head = c0 / C;
    const float al = alpha[edge * H + head];
    const float4 hv = *(const float4*)(h + (size_t)s * HC + c0);
    float* o = out + (size_t)d * HC + c0;
    atomicAdd(o + 0, hv.x * al);
    atomicAdd(o + 1, hv.y * al);
    atomicAdd(o + 2, hv.z * al);
    atomicAdd(o + 3, hv.w * al);
}

__global__ void bias_act_kernel(const float* __restrict__ in, const float* __restrict__ b,
                                float* __restrict__ out, long long n, int HC, int do_relu) {
    const long long i = (long long)blockIdx.x * blockDim.x + threadIdx.x;
    if (i >= n) return;
    float v = in[i] + b[i % HC];
    if (do_relu) v = fmaxf(v, 0.f);
    out[i] = v;
}

// ---------- decode: one wave32 per positive edge, dot over D channels ----------
__global__ void dot_kernel(const float* __restrict__ z, const long long* __restrict__ pos,
                           float* __restrict__ out, int Ep, int D) {
    const int w    = (blockIdx.x * blockDim.x + threadIdx.x) >> 5;
    const int lane = threadIdx.x & 31;
    if (w >= Ep) return;
    const int a = (int)pos[w];
    const int b = (int)pos[Ep + w];
    const float* za = z + (size_t)a * D;
    const float* zb = z + (size_t)b * D;
    float s = 0.f;
    for (int c = lane * 4; c < D; c += 128) {
        const float4 x = *(const float4*)(za + c);
        const float4 y = *(const float4*)(zb + c);
        s += x.x * y.x + x.y * y.y + x.z * y.z + x.w * y.w;
    }
#pragma unroll
    for (int off = 16; off > 0; off >>= 1) s += __shfl_down(s, off, 32);
    if (lane == 0) out[w] = s;
}

// ---------- host ----------

static inline unsigned blocks1d(long long n, int bs) { return (unsigned)((n + bs - 1) / bs); }

extern "C" void kernel_launch(void* const* d_in, const int* in_sizes, int n_in,
                              void* d_out, int out_size, void* d_ws, size_t ws_size,
                              hipStream_t stream) {
    const float*     x   = (const float*)d_in[0];
    const long long* ei  = (const long long*)d_in[1];   // int64 per reference
    const long long* pos = (const long long*)d_in[2];
    const float*     W1  = (const float*)d_in[3];
    const float*     a1s = (const float*)d_in[4];
    const float*     a1d = (const float*)d_in[5];
    const float*     b1  = (const float*)d_in[6];
    const float*     W2  = (const float*)d_in[7];
    const float*     a2s = (const float*)d_in[8];
    const float*     a2d = (const float*)d_in[9];
    const float*     b2  = (const float*)d_in[10];

    const int H   = 2;
    const int HC1 = in_sizes[6];            // H*Hd = 256
    const int HC2 = in_sizes[9];            // H*Co = 128
    const int C1  = HC1 / H;
    const int C2  = HC2 / H;
    const int Fin = in_sizes[3] / HC1;      // 128
    const int N   = in_sizes[0] / Fin;      // 50000
    const int E   = in_sizes[1] / 2;        // 800000
    const int Ep  = in_sizes[2] / 2;        // 100000
    const int Etot = E + N;                 // + self loops

    // workspace carve-out (256B aligned)
    char* ws = (char*)d_ws;
    size_t off = 0;
    auto carve = [&](size_t bytes) -> char* {
        char* p = ws + off;
        off += (bytes + 255) & ~(size_t)255;
        return p;
    };
    float*    h1   = (float*)carve((size_t)N * HC1 * 4);
    float*    out1 = (float*)carve((size_t)N * HC1 * 4);
    float*    h2   = (float*)carve((size_t)N * HC2 * 4);
    float*    out2 = (float*)carve((size_t)N * HC2 * 4);
    float*    as1  = (float*)carve((size_t)N * H * 4);
    float*    ad1  = (float*)carve((size_t)N * H * 4);
    unsigned* m1   = (unsigned*)carve((size_t)N * H * 4);
    float*    s1   = (float*)carve((size_t)N * H * 4);
    float*    as2  = (float*)carve((size_t)N * H * 4);
    float*    ad2  = (float*)carve((size_t)N * H * 4);
    unsigned* m2   = (unsigned*)carve((size_t)N * H * 4);
    float*    s2   = (float*)carve((size_t)N * H * 4);
    float*    exb  = (float*)carve((size_t)Etot * H * 4);
    (void)ws_size; (void)n_in;

    const int BS = 256;

    // ---- init accumulators (every call: harness does not re-poison) ----
    const unsigned ordNegInf = 0x007FFFFFu;  // f2ord(-inf) = ~0xFF800000
    fill_u32_kernel<<<blocks1d((long long)N * HC1, BS), BS, 0, stream>>>((unsigned*)out1, 0u, (long long)N * HC1);
    fill_u32_kernel<<<blocks1d((long long)N * HC2, BS), BS, 0, stream>>>((unsigned*)out2, 0u, (long long)N * HC2);
    fill_u32_kernel<<<blocks1d((long long)N * H, BS), BS, 0, stream>>>((unsigned*)s1, 0u, (long long)N * H);
    fill_u32_kernel<<<blocks1d((long long)N * H, BS), BS, 0, stream>>>((unsigned*)s2, 0u, (long long)N * H);
    fill_u32_kernel<<<blocks1d((long long)N * H, BS), BS, 0, stream>>>(m1, ordNegInf, (long long)N * H);
    fill_u32_kernel<<<blocks1d((long long)N * H, BS), BS, 0, stream>>>(m2, ordNegInf, (long long)N * H);

    // ---- layer 1 ----
    {
        const int tiles = (N / 16) * (HC1 / 16);
        wmma_gemm_f32_kernel<<<blocks1d(tiles, 8), BS, 0, stream>>>(x, W1, h1, N, Fin, HC1);
        alpha_kernel<<<blocks1d((long long)N * H * 32, BS), BS, 0, stream>>>(h1, a1s, a1d, as1, ad1, N, H, C1);
        edge_max_kernel<<<blocks1d((long long)Etot * H, BS), BS, 0, stream>>>(as1, ad1, ei, m1, E, Etot, H);
        decode_max_kernel<<<blocks1d((long long)N * H, BS), BS, 0, stream>>>(m1, N * H);
        edge_exp_kernel<<<blocks1d((long long)Etot * H, BS), BS, 0, stream>>>(as1, ad1, ei, (const float*)m1, s1, exb, E, Etot, H);
        edge_norm_kernel<<<blocks1d((long long)Etot * H, BS), BS, 0, stream>>>(ei, s1, exb, E, Etot, H);
        agg_kernel<<<blocks1d((long long)Etot * (HC1 / 4), BS), BS, 0, stream>>>(h1, exb, ei, out1, E, Etot, H, C1, HC1);
        // h1 <- relu(out1 + b1)  (h1 buffer is free after aggregation)
        bias_act_kernel<<<blocks1d((long long)N * HC1, BS), BS, 0, stream>>>(out1, b1, h1, (long long)N * HC1, HC1, 1);
    }

    // ---- layer 2 ----
    {
        const int tiles = (N / 16) * (HC2 / 16);
        wmma_gemm_f32_kernel<<<blocks1d(tiles, 8), BS, 0, stream>>>(h1, W2, h2, N, HC1, HC2);
        alpha_kernel<<<blocks1d((long long)N * H * 32, BS), BS, 0, stream>>>(h2, a2s, a2d, as2, ad2, N, H, C2);
        edge_max_kernel<<<blocks1d((long long)Etot * H, BS), BS, 0, stream>>>(as2, ad2, ei, m2, E, Etot, H);
        decode_max_kernel<<<blocks1d((long long)N * H, BS), BS, 0, stream>>>(m2, N * H);
        edge_exp_kernel<<<blocks1d((long long)Etot * H, BS), BS, 0, stream>>>(as2, ad2, ei, (const float*)m2, s2, exb, E, Etot, H);
        edge_norm_kernel<<<blocks1d((long long)Etot * H, BS), BS, 0, stream>>>(ei, s2, exb, E, Etot, H);
        agg_kernel<<<blocks1d((long long)Etot * (HC2 / 4), BS), BS, 0, stream>>>(h2, exb, ei, out2, E, Etot, H, C2, HC2);
        // z = out2 + b2 (no relu), in place
        bias_act_kernel<<<blocks1d((long long)N * HC2, BS), BS, 0, stream>>>(out2, b2, out2, (long long)N * HC2, HC2, 0);
    }

    // ---- decode ----
    dot_kernel<<<blocks1d((long long)Ep * 32, BS), BS, 0, stream>>>(out2, pos, (float*)d_out, Ep, HC2);
}